// SurfaceNormalOptimizer_23759759081641
// MI455X (gfx1250) — compile-verified
//
#include <hip/hip_runtime.h>
#include <hip/hip_bf16.h>
#include <math.h>

// Problem constants (match reference: B=8, H=384, W=1280, SCALE=2)
#define BB 8
#define HH 384
#define WW 1280
#define LH 96   // H / 4
#define LW 320  // W / 4

typedef __attribute__((ext_vector_type(2))) float v2f;
typedef __attribute__((ext_vector_type(8))) float v8f;

// Each wave handles 16 cells = a 4-row x 64-col pixel band.
// Per cell: out[p] = depth * exp( sum_k C[p][k] * L[k] ),
// C = 16x32 constant coefficient matrix derived from the 15 linearized paths.
// D(16x16) = C(16x32) x L(32x16) via 8x V_WMMA_F32_16X16X4_F32.

__global__ __launch_bounds__(256) void
SurfaceNormalOptimizer_kernel(const float* __restrict__ depth,
                              const float* __restrict__ ang,
                              const float* __restrict__ intr,
                              float* __restrict__ out)
{
    __shared__ float sC[16][32];      // coefficient matrix [pos][k]
    __shared__ float sL[8][32][16];   // per-wave L values  [k][cell]

    const int tid  = threadIdx.x;
    const int lane = tid & 31;
    const int w    = tid >> 5;   // wave in block (0..7)
    const int l2   = lane & 15;
    const int hi   = lane >> 4;

    // ---- Phase 0: derive coefficient matrix by replaying the 15 paths ----
    // pos(dy,dx) = (dy+2)*4 + (dx+2);  k(ch,ly,lx) = ch*16 + pos(ly,lx)
    if (tid == 0) {
        const int   pA[15] = {10,10, 9,10, 6, 6, 5,10,14,14,13, 6, 2, 2, 1};
        const int   kA[15] = { 9,10, 8,22, 5, 6, 4,26,13,14,12,18, 1, 2, 0};
        const float sA[15] = {-1, 1,-1,-1,-1, 1,-1, 1,-1, 1,-1,-1,-1, 1,-1};
        const int   pB[15] = {10,10, 9,10, 9,11, 8,10, 9,11, 8, 6, 5, 7, 4};
        const int   kB[15] = { 9,10, 8,22,21,23,20,26,25,27,24,18,17,19,16};
        const float sB[15] = {-1, 1,-1,-1,-1,-1,-1, 1, 1, 1, 1,-1,-1,-1,-1};
        const int   pT[15] = { 9,11, 8, 6, 5, 7, 4,14,13,15,12, 2, 1, 3, 0};
        for (int p = 0; p < 16; ++p)
            for (int k = 0; k < 32; ++k) sC[p][k] = 0.0f;
        for (int i = 0; i < 15; ++i) {
            for (int k = 0; k < 32; ++k)
                sC[pT[i]][k] = 0.5f * (sC[pA[i]][k] + sC[pB[i]][k]);
            sC[pT[i]][kA[i]] += 0.5f * sA[i];
            sC[pT[i]][kB[i]] += 0.5f * sB[i];
        }
    }

    // ---- Wave tile decomposition ----
    const int wid  = blockIdx.x * 8 + w;     // global wave id
    const int cw   = wid % (LW / 16);        // 16-cell group along x (0..19)
    const int rowb = wid / (LW / 16);
    const int row  = rowb % LH;              // cell row
    const int b    = rowb / LH;              // batch

    const int x0 = cw * 64;                  // pixel tile origin
    const int y0 = row * 4;

    const float fx = intr[b * 9 + 0];
    const float bx = intr[b * 9 + 2];
    const float fy = intr[b * 9 + 4];
    const float by = intr[b * 9 + 5];

    const float* angh_p = ang + (size_t)(b * 2 + 0) * HH * WW;
    const float* angv_p = ang + (size_t)(b * 2 + 1) * HH * WW;

    // ---- Phase 1: coalesced ang loads, compute logh/logv, stash in LDS ----
    #pragma unroll
    for (int i = 0; i < 8; ++i) {
        const int t   = lane + 32 * i;       // 256 pixels per wave tile
        const int r   = t >> 6;              // row in band (0..3)
        const int col = t & 63;              // col in band (0..63)
        const int x = x0 + col, y = y0 + r;
        const size_t off = (size_t)y * WW + x;
        const float ah = angh_p[off];
        const float av = angv_p[off];

        const float xx = (float)x, yy = (float)y;
        const float ry = (yy - by) / fy;
        const float rx = (xx - bx) / fx;

        // logh: a1=ry^2+1, b1=-rx, b2=-(x+1-bx)/fx, a3=sin, b3=-cos
        const float a1 = ry * ry + 1.0f;
        const float b1 = -rx;
        const float b2 = -(xx + 1.0f - bx) / fx;
        float sh, chh; __sincosf(ah, &sh, &chh);
        const float t1 = fabsf(sh * b1 + a1 * chh);   // a3*b1 - a1*b3
        const float t2 = fabsf(sh * b2 + a1 * chh);
        float logh = __logf(fmaxf(t1, 1e-6f)) - __logf(fmaxf(t2, 1e-6f));
        logh = fminf(fmaxf(logh, -10.0f), 10.0f);

        // logv: u1=rx^2+1, v1=-ry, v2=-(y+1-by)/fy
        const float u1 = rx * rx + 1.0f;
        const float v1 = -ry;
        const float v2 = -(yy + 1.0f - by) / fy;
        float sv, cvv; __sincosf(av, &sv, &cvv);
        const float t3 = fabsf(sv * v1 + u1 * cvv);
        const float t4 = fabsf(sv * v2 + u1 * cvv);
        float logv = __logf(fmaxf(t3, 1e-6f)) - __logf(fmaxf(t4, 1e-6f));
        logv = fminf(fmaxf(logv, -10.0f), 10.0f);

        const int cell = col >> 2, c = col & 3;
        const int pos  = r * 4 + c;
        sL[w][pos][cell]      = logh;   // k = 0*16 + pos
        sL[w][16 + pos][cell] = logv;   // k = 1*16 + pos
    }

    __syncthreads();   // publish sC (sL is strictly wave-local)

    // ---- Phase 2: 8 x WMMA f32 16x16x4, K split into 4-wide chunks ----
    // 32-bit A 16x4 layout: lane half hi selects K pair: VGPRj -> K = 4q + 2*hi + j, M = l2
    // B 4x16 symmetric: VGPRj -> K = 4q + 2*hi + j, N = l2
    v2f afr[8];
    #pragma unroll
    for (int q = 0; q < 8; ++q) {
        afr[q].x = sC[l2][4 * q + 2 * hi + 0];
        afr[q].y = sC[l2][4 * q + 2 * hi + 1];
    }
    v8f acc = {};
    #pragma unroll
    for (int q = 0; q < 8; ++q) {
        v2f bfr;
        bfr.x = sL[w][4 * q + 2 * hi + 0][l2];
        bfr.y = sL[w][4 * q + 2 * hi + 1][l2];
        acc = __builtin_amdgcn_wmma_f32_16x16x4_f32(
            /*neg_a=*/false, afr[q], /*neg_b=*/false, bfr,
            /*c_mod=*/(short)0, acc, /*reuse_a=*/false, /*reuse_b=*/false);
    }

    // ---- Phase 3: out = depth * exp(acc); stage via LDS for coalesced stores
    const float dv = depth[((size_t)b * LH + row) * LW + cw * 16 + l2];

    // D layout: VGPR i, lanes0-15 -> M=i, lanes16-31 -> M=i+8; N = l2
    #pragma unroll
    for (int i = 0; i < 8; ++i) {
        const int pos = i + 8 * hi;
        sL[w][pos][l2] = dv * __expf(acc[i]);
    }

    float* outp = out + (size_t)b * HH * WW;
    #pragma unroll
    for (int i = 0; i < 8; ++i) {
        const int t   = lane + 32 * i;
        const int r   = t >> 6;
        const int col = t & 63;
        const int cell = col >> 2, c = col & 3;
        const int pos  = r * 4 + c;
        outp[(size_t)(y0 + r) * WW + x0 + col] = sL[w][pos][cell];
    }
}

extern "C" void kernel_launch(void* const* d_in, const int* in_sizes, int n_in,
                              void* d_out, int out_size, void* d_ws, size_t ws_size,
                              hipStream_t stream) {
    (void)in_sizes; (void)n_in; (void)d_ws; (void)ws_size; (void)out_size;
    const float* depth = (const float*)d_in[0];   // (B,1,96,320) f32
    const float* ang   = (const float*)d_in[1];   // (B,2,384,1280) f32
    const float* intr  = (const float*)d_in[2];   // (B,3,3) f32
    // d_in[3] = scale (==2), baked into the kernel constants.
    float* outp = (float*)d_out;                  // (B,1,384,1280) f32

    const int cells  = BB * LH * LW;              // 245760
    const int waves  = cells / 16;                // 15360
    const int blocks = waves / 8;                 // 1920
    SurfaceNormalOptimizer_kernel<<<blocks, 256, 0, stream>>>(depth, ang, intr, outp);
}